// PagedAttention_84035330113923
// MI455X (gfx1250) — compile-verified
//
#include <hip/hip_runtime.h>

// Paged attention decode for MI455X (gfx1250, wave32).
//
// Roofline: K/V stream = 2*B*T*H*D*4B ~= 536 MB -> ~23us floor at 23.3 TB/s;
// AI ~0.5 flop/byte => purely HBM-bound, fp32 end-to-end (downcasting cannot
// reduce the fp32 cache bytes). V_WMMA_F32_16X16X4_F32 does both QK^T and P*V
// with the vector operand broadcast across the 16-wide dimension.
//
// Latency hiding: saturating 23.3 TB/s needs ~BW*latency ~ 23 MB in flight.
// One WG per (b,h) (2048 waves) only sustains ~8 MB, so we flash-decode:
// SPLITS=4 context partitions -> 1024 WGs / 8192 waves, per-split partials
// (o, m, l) in workspace, then a cheap log-sum-exp combine kernel.

#define NUM_HEADS 16
#define HEAD_DIM 128
#define BLOCK_SZ 16
#define BLOCKS_PER_SEQ 128
#define BATCH 16
#define NUM_BH (BATCH * NUM_HEADS)
#define WAVES 8
#define SPLITS 4
#define BLOCKS_PER_SPLIT (BLOCKS_PER_SEQ / SPLITS)  // 32
#define NEG_BIG -1.0e30f

typedef __attribute__((ext_vector_type(2))) float v2f;
typedef __attribute__((ext_vector_type(8))) float v8f;

// ---------------- Stage 1: per-split partial attention ----------------
__global__ __launch_bounds__(256) void paged_attn_partial(
    const float* __restrict__ q,     // [B, H, D]
    const float* __restrict__ kc,    // [NB, BS, H, D]
    const float* __restrict__ vc,    // [NB, BS, H, D]
    const int*   __restrict__ bt,    // [B, BLOCKS_PER_SEQ]
    const int*   __restrict__ cl,    // [B]
    float*       __restrict__ p_out, // [SPLITS, B*H, D]  unnormalized numerator
    float*       __restrict__ p_m,   // [SPLITS, B*H]     split max
    float*       __restrict__ p_l)   // [SPLITS, B*H]     split sum (in m-frame)
{
    __shared__ float s_out[WAVES][HEAD_DIM];
    __shared__ float s_m[WAVES];
    __shared__ float s_l[WAVES];

    const int bh    = blockIdx.x & (NUM_BH - 1);
    const int split = blockIdx.x >> 8;           // NUM_BH == 256
    const int b     = bh >> 4;
    const int h     = bh & 15;
    const int tid   = threadIdx.x;
    const int wave  = tid >> 5;
    const int lane  = tid & 31;
    const int lo    = lane & 15;
    const bool hi   = lane >= 16;
    const int hi2   = hi ? 2 : 0;                // k-slot offset for this lane half
    const int hi8   = hi ? 8 : 0;                // M-row offset in C/D layout

    const int ctx       = cl[b];
    const int nblocks   = (ctx + BLOCK_SZ - 1) / BLOCK_SZ;
    const int blk_begin = split * BLOCKS_PER_SPLIT;
    const int blk_end   = min(nblocks, blk_begin + BLOCKS_PER_SPLIT);
    const float scale   = 0.08838834764831843f;  // 1/sqrt(128)
    const size_t tstride = (size_t)NUM_HEADS * HEAD_DIM;  // floats per timestep

    // Preload query pairs: qv[i] = {q[4i+hi2], q[4i+hi2+1]} matches the
    // per-lane B-matrix slot layout for a column-broadcast operand.
    const float* qp = q + (size_t)bh * HEAD_DIM;
    v2f qv[32];
#pragma unroll
    for (int i = 0; i < 32; ++i)
        qv[i] = *(const v2f*)(qp + 4 * i + hi2);

    float m = NEG_BIG;
    float l = 0.0f;
    v8f acc[8];  // out[128] accumulators (8 N-tiles of 16 dims), replicated over M
#pragma unroll
    for (int dt = 0; dt < 8; ++dt) {
        v8f z = {};
        acc[dt] = z;
    }

    for (int bi = blk_begin + wave; bi < blk_end; bi += WAVES) {
        const int phys = bt[b * BLOCKS_PER_SEQ + bi];
        const float* kb = kc + ((size_t)phys * BLOCK_SZ * NUM_HEADS + h) * HEAD_DIM;
        const float* vb = vc + ((size_t)phys * BLOCK_SZ * NUM_HEADS + h) * HEAD_DIM;

        // ---- QK^T: scores[t] = sum_d K[t,d]*q[d] for 16 timesteps ----
        // A = K tile (M=t, K=dims), B = q broadcast over N.
        v8f sacc = {};
        const float* krow = kb + (size_t)lo * tstride + hi2;  // row t = lo
#pragma unroll
        for (int i = 0; i < 32; ++i) {
            v2f a = *(const v2f*)(krow + 4 * i);
            sacc = __builtin_amdgcn_wmma_f32_16x16x4_f32(
                false, a, false, qv[i], (short)0, sacc, false, false);
        }

        // ---- online softmax over this 16-timestep tile ----
        float sv[8], pv[8];
        float tmax = NEG_BIG;
#pragma unroll
        for (int j = 0; j < 8; ++j) {
            const int tg = bi * BLOCK_SZ + j + hi8;
            float s = sacc[j] * scale;
            s = (tg < ctx) ? s : NEG_BIG;
            sv[j] = s;
            tmax = fmaxf(tmax, s);
        }
        tmax = fmaxf(tmax, __shfl_xor(tmax, 16, 32));  // merge M=0-7 / M=8-15 halves
        const float mnew  = fmaxf(m, tmax);
        const float alpha = __expf(m - mnew);
        float psum = 0.0f;
#pragma unroll
        for (int j = 0; j < 8; ++j) {
            const int tg = bi * BLOCK_SZ + j + hi8;
            const float p = (tg < ctx) ? __expf(sv[j] - mnew) : 0.0f;
            pv[j] = p;
            psum += p;
        }
        psum += __shfl_xor(psum, 16, 32);
        l = l * alpha + psum;
        m = mnew;

        // Broadcast the 16 probabilities to wave-uniform values.
        float ps[16];
#pragma unroll
        for (int j = 0; j < 8; ++j) {
            ps[j]     = __shfl(pv[j], 0, 32);   // rows 0..7 live in lanes 0-15
            ps[j + 8] = __shfl(pv[j], 16, 32);  // rows 8..15 live in lanes 16-31
        }

        // Rescale accumulators by alpha.
#pragma unroll
        for (int dt = 0; dt < 8; ++dt)
#pragma unroll
            for (int j = 0; j < 8; ++j)
                acc[dt][j] *= alpha;

        // ---- P*V: out[d] += sum_t p[t]*V[t,d] ----
        // A = p broadcast over M (K=4 timesteps), B = V tile (K=t, N=dims).
#pragma unroll
        for (int t0 = 0; t0 < 16; t0 += 4) {
            v2f a;
            a[0] = hi ? ps[t0 + 2] : ps[t0 + 0];
            a[1] = hi ? ps[t0 + 3] : ps[t0 + 1];
            const float* vrow = vb + (size_t)(t0 + hi2) * tstride + lo;
#pragma unroll
            for (int dt = 0; dt < 8; ++dt) {
                v2f bv;
                bv[0] = vrow[dt * 16];            // V[t0+hi2,   dt*16+lo]
                bv[1] = vrow[tstride + dt * 16];  // V[t0+hi2+1, dt*16+lo]
                acc[dt] = __builtin_amdgcn_wmma_f32_16x16x4_f32(
                    false, a, false, bv, (short)0, acc[dt], false, false);
            }
        }
    }

    // ---- per-wave partials to LDS (rows replicated over M -> take comp 0) ----
    if (lane < 16) {
#pragma unroll
        for (int dt = 0; dt < 8; ++dt)
            s_out[wave][dt * 16 + lane] = acc[dt][0];
    }
    if (lane == 0) { s_m[wave] = m; s_l[wave] = l; }
    __syncthreads();

    // ---- combine the 8 waves of this split; emit unnormalized partial ----
    if (tid < HEAD_DIM) {
        float M = NEG_BIG;
#pragma unroll
        for (int w = 0; w < WAVES; ++w) M = fmaxf(M, s_m[w]);
        float L = 0.0f, o = 0.0f;
#pragma unroll
        for (int w = 0; w < WAVES; ++w) {
            const float e = __expf(s_m[w] - M);
            L += s_l[w] * e;
            o += s_out[w][tid] * e;
        }
        const int pidx = split * NUM_BH + bh;
        p_out[(size_t)pidx * HEAD_DIM + tid] = o;
        if (tid == 0) { p_m[pidx] = M; p_l[pidx] = L; }
    }
}

// ---------------- Stage 2: log-sum-exp combine over splits ----------------
__global__ __launch_bounds__(HEAD_DIM) void paged_attn_combine(
    const float* __restrict__ p_out,
    const float* __restrict__ p_m,
    const float* __restrict__ p_l,
    float*       __restrict__ out)
{
    const int bh = blockIdx.x;
    const int d  = threadIdx.x;

    float M = NEG_BIG;
#pragma unroll
    for (int s = 0; s < SPLITS; ++s) M = fmaxf(M, p_m[s * NUM_BH + bh]);
    float L = 0.0f, o = 0.0f;
#pragma unroll
    for (int s = 0; s < SPLITS; ++s) {
        const int pidx = s * NUM_BH + bh;
        const float e = __expf(p_m[pidx] - M);
        L += p_l[pidx] * e;
        o += p_out[(size_t)pidx * HEAD_DIM + d] * e;
    }
    out[(size_t)bh * HEAD_DIM + d] = o / fmaxf(L, 1e-30f);
}

extern "C" void kernel_launch(void* const* d_in, const int* in_sizes, int n_in,
                              void* d_out, int out_size, void* d_ws, size_t ws_size,
                              hipStream_t stream) {
    const float* q  = (const float*)d_in[0];
    const float* kc = (const float*)d_in[1];
    const float* vc = (const float*)d_in[2];
    const int*   bt = (const int*)d_in[3];
    const int*   cl = (const int*)d_in[4];
    // d_in[5] = max_context_len (compile-time constant 2048 here)
    float* out = (float*)d_out;

    // Workspace layout: [SPLITS*256*128] out partials, then m[1024], l[1024].
    float* p_out = (float*)d_ws;
    float* p_m   = p_out + (size_t)SPLITS * NUM_BH * HEAD_DIM;
    float* p_l   = p_m + SPLITS * NUM_BH;

    dim3 grid1(NUM_BH * SPLITS);  // 1024 WGs: (split, b, h)
    dim3 block1(32 * WAVES);      // 8 waves (wave32)
    paged_attn_partial<<<grid1, block1, 0, stream>>>(q, kc, vc, bt, cl,
                                                     p_out, p_m, p_l);

    dim3 grid2(NUM_BH);
    dim3 block2(HEAD_DIM);
    paged_attn_combine<<<grid2, block2, 0, stream>>>(p_out, p_m, p_l, out);
}